// FAGCN_66005057405291
// MI455X (gfx1250) — compile-verified
//
#include <hip/hip_runtime.h>
#include <math.h>

#define N_NODES   50000
#define N_EDGES   1600000
#define NUM_FEAT  512
#define HIDDEN    128
#define NUM_CLASS 40
#define NUM_LAYERS 4
#define EPS_VAL   0.3f
#define NODE_TILES (N_NODES / 16)          // 3125 exactly
#define SCAN_B    256
#define SCAN_NB   ((N_NODES + SCAN_B - 1) / SCAN_B)   // 196
#define LDS_PITCH 516                      // 512 + 4 -> A-lanes hit distinct LDS banks

typedef float v2f __attribute__((ext_vector_type(2)));
typedef float v8f __attribute__((ext_vector_type(8)));

// ---------------------------------------------------------------- CSR build
__global__ __launch_bounds__(256) void k_count(const int* __restrict__ ei,
                                               int* __restrict__ cnt) {
    int e = blockIdx.x * 256 + threadIdx.x;
    if (e < N_EDGES) atomicAdd(&cnt[ei[N_EDGES + e]], 1);      // dst = ei[1][e]
}

// inclusive scan of 256-element chunks + per-block totals
__global__ __launch_bounds__(SCAN_B) void k_scan1(const int* __restrict__ cnt,
                                                  int* __restrict__ partial,
                                                  int* __restrict__ blocksum) {
    __shared__ int sh[SCAN_B];
    const int tid = threadIdx.x;
    const int g   = blockIdx.x * SCAN_B + tid;
    sh[tid] = (g < N_NODES) ? cnt[g] : 0;
    __syncthreads();
#pragma unroll
    for (int off = 1; off < SCAN_B; off <<= 1) {
        int t = (tid >= off) ? sh[tid - off] : 0;
        __syncthreads();
        sh[tid] += t;
        __syncthreads();
    }
    if (g < N_NODES) partial[g] = sh[tid];
    if (tid == SCAN_B - 1) blocksum[blockIdx.x] = sh[tid];
}

__global__ __launch_bounds__(SCAN_B) void k_scan2(int* __restrict__ blocksum) {
    __shared__ int sh[SCAN_B];
    const int tid = threadIdx.x;
    sh[tid] = (tid < SCAN_NB) ? blocksum[tid] : 0;
    __syncthreads();
#pragma unroll
    for (int off = 1; off < SCAN_B; off <<= 1) {
        int t = (tid >= off) ? sh[tid - off] : 0;
        __syncthreads();
        sh[tid] += t;
        __syncthreads();
    }
    if (tid < SCAN_NB) blocksum[tid] = sh[tid];    // inclusive block sums
}

__global__ __launch_bounds__(SCAN_B) void k_scan3(const int* __restrict__ partial,
                                                  const int* __restrict__ blocksum,
                                                  int* __restrict__ rowptr) {
    const int g = blockIdx.x * SCAN_B + threadIdx.x;
    if (g >= N_NODES) return;
    const int boff = (blockIdx.x > 0) ? blocksum[blockIdx.x - 1] : 0;
    rowptr[g + 1] = partial[g] + boff;             // inclusive scan -> rowptr[i+1]
    if (g == 0) rowptr[0] = 0;
}

__global__ __launch_bounds__(256) void k_dis_cursor(const int* __restrict__ cnt,
                                                    const int* __restrict__ rowptr,
                                                    float* __restrict__ dis,
                                                    int* __restrict__ cursor) {
    int i = blockIdx.x * 256 + threadIdx.x;
    if (i >= N_NODES) return;
    dis[i]    = rsqrtf((float)cnt[i] + 1.0f);      // +1 = self loop; deg >= 1 always
    cursor[i] = rowptr[i];
}

__global__ __launch_bounds__(256) void k_fill(const int* __restrict__ ei,
                                              const float* __restrict__ dis,
                                              int* __restrict__ cursor,
                                              int* __restrict__ csr_src,
                                              int* __restrict__ csr_dst,
                                              float* __restrict__ csr_norm) {
    int e = blockIdx.x * 256 + threadIdx.x;
    if (e >= N_EDGES) return;
    const int j = ei[e];                 // source
    const int i = ei[N_EDGES + e];       // destination
    const int pos = atomicAdd(&cursor[i], 1);
    csr_src[pos]  = j;
    csr_dst[pos]  = i;
    csr_norm[pos] = dis[j] * dis[i];     // layer-invariant part of the coefficient
}

// ------------------------------------------------- GEMM1: h = relu(x*W1^T + b1)
// grid = 3125 node tiles, block = 128 (4 waves). x-tile (16x512) staged in LDS,
// shared by all 4 waves; wave w covers hidden cols [w*32, w*32+32).
__global__ __launch_bounds__(128) void k_gemm1(const float* __restrict__ x,
                                               const float* __restrict__ W1,
                                               const float* __restrict__ b1,
                                               float* __restrict__ h) {
    __shared__ float tile[16 * LDS_PITCH];         // 33 KB, padded rows
    const int nt   = blockIdx.x;
    const int wave = threadIdx.x >> 5;
    const int lane = threadIdx.x & 31;
    const int l15  = lane & 15;
    const int hi   = lane >> 4;        // K-pair select: 0 -> {k,k+1}, 1 -> {k+2,k+3}
    const int col0 = wave * 32;

    // cooperative coalesced load of the 16x512 x tile (2048 float4s / 128 threads)
#pragma unroll
    for (int it = 0; it < 16; ++it) {
        const int fi  = threadIdx.x + 128 * it;    // float4 index 0..2047
        const int row = fi >> 7;                   // 128 float4 per row
        const int c4  = fi & 127;
        const float4 v = *(const float4*)(x + (size_t)(nt * 16 + row) * NUM_FEAT + c4 * 4);
        *(float4*)&tile[row * LDS_PITCH + c4 * 4] = v;
    }
    __syncthreads();

    const float* w0  = W1 + (size_t)(col0 +  0 + l15) * NUM_FEAT;
    const float* w1p = W1 + (size_t)(col0 + 16 + l15) * NUM_FEAT;

    v8f acc0 = {}, acc1 = {};
    for (int k = 0; k < NUM_FEAT; k += 4) {
        const int kb = k + 2 * hi;
        v2f a  = *(const v2f*)&tile[l15 * LDS_PITCH + kb];   // A: 16x4 f32 (bank-clean)
        v2f b0 = *(const v2f*)(w0  + kb);                    // B: 4x16 f32 tiles
        v2f b1v = *(const v2f*)(w1p + kb);
        acc0 = __builtin_amdgcn_wmma_f32_16x16x4_f32(false, a, false, b0,  (short)0, acc0, false, false);
        acc1 = __builtin_amdgcn_wmma_f32_16x16x4_f32(false, a, false, b1v, (short)0, acc1, false, false);
    }

    const int c0 = col0 + l15;
    const float bb0 = b1[c0], bb1 = b1[c0 + 16];
#pragma unroll
    for (int r = 0; r < 8; ++r) {
        const size_t orow = (size_t)(nt * 16 + r + hi * 8) * HIDDEN;
        h[orow + c0     ] = fmaxf(acc0[r] + bb0, 0.0f);
        h[orow + c0 + 16] = fmaxf(acc1[r] + bb1, 0.0f);
    }
}

// ---------------------------------------------- per-node attention dots al, ar
__global__ __launch_bounds__(256) void k_att(const float* __restrict__ h,
                                             const float* __restrict__ attl,
                                             const float* __restrict__ attr,
                                             float* __restrict__ al,
                                             float* __restrict__ ar) {
    const int n    = (blockIdx.x * 256 + threadIdx.x) >> 5;
    const int lane = threadIdx.x & 31;
    if (n >= N_NODES) return;
    const float4 hv = *(const float4*)(h + (size_t)n * HIDDEN + lane * 4);
    const float4 lv = *(const float4*)(attl + lane * 4);
    const float4 rv = *(const float4*)(attr + lane * 4);
    float sl = hv.x * lv.x + hv.y * lv.y + hv.z * lv.z + hv.w * lv.w;
    float sr = hv.x * rv.x + hv.y * rv.y + hv.z * rv.z + hv.w * rv.w;
#pragma unroll
    for (int m = 16; m >= 1; m >>= 1) {
        sl += __shfl_xor(sl, m, 32);
        sr += __shfl_xor(sr, m, 32);
    }
    if (lane == 0) { al[n] = sl; ar[n] = sr; }
}

// --------------- per-edge coefficient (one tanh per edge, fully coalesced)
__global__ __launch_bounds__(256) void k_coef(const int* __restrict__ csr_src,
                                              const int* __restrict__ csr_dst,
                                              const float* __restrict__ csr_norm,
                                              const float* __restrict__ al,
                                              const float* __restrict__ ar,
                                              float* __restrict__ coef) {
    int k = blockIdx.x * 256 + threadIdx.x;
    if (k >= N_EDGES) return;
    coef[k] = tanhf(al[csr_src[k]] + ar[csr_dst[k]]) * csr_norm[k];
}

// --------- pull-mode aggregation: one wave per destination node, no atomics.
// hout[i] = sum_j coef(j,i)*h[j] + selfloop(i)*h[i] + EPS*raw[i]
// lane owns 4 contiguous features -> one b128 load per edge-row.
__global__ __launch_bounds__(256) void k_agg(const int* __restrict__ rowptr,
                                             const int* __restrict__ csr_src,
                                             const float* __restrict__ coef,
                                             const float* __restrict__ h,
                                             const float* __restrict__ raw,
                                             const float* __restrict__ al,
                                             const float* __restrict__ ar,
                                             const float* __restrict__ dis,
                                             float* __restrict__ hout) {
    const int i    = (blockIdx.x * 256 + threadIdx.x) >> 5;
    const int lane = threadIdx.x & 31;
    if (i >= N_NODES) return;

    const float d  = dis[i];
    const float cs = tanhf(al[i] + ar[i]) * d * d;         // self-loop coefficient
    const float4 hv = ((const float4*)(h   + (size_t)i * HIDDEN))[lane];
    const float4 rv = ((const float4*)(raw + (size_t)i * HIDDEN))[lane];
    float a0 = cs * hv.x + EPS_VAL * rv.x;
    float a1 = cs * hv.y + EPS_VAL * rv.y;
    float a2 = cs * hv.z + EPS_VAL * rv.z;
    float a3 = cs * hv.w + EPS_VAL * rv.w;

    const int kend = rowptr[i + 1];
    int k = rowptr[i];
    for (; k + 2 <= kend; k += 2) {                        // 2-edge unroll for ILP
        __builtin_prefetch(csr_src + k + 32, 0, 1);        // global_prefetch_b8
        const int   ja = csr_src[k];
        const int   jb = csr_src[k + 1];
        const float ca = coef[k];
        const float cb = coef[k + 1];
        const float4 va = ((const float4*)(h + (size_t)ja * HIDDEN))[lane];
        const float4 vb = ((const float4*)(h + (size_t)jb * HIDDEN))[lane];
        a0 += ca * va.x + cb * vb.x;
        a1 += ca * va.y + cb * vb.y;
        a2 += ca * va.z + cb * vb.z;
        a3 += ca * va.w + cb * vb.w;
    }
    if (k < kend) {
        const int   j = csr_src[k];
        const float c = coef[k];
        const float4 v = ((const float4*)(h + (size_t)j * HIDDEN))[lane];
        a0 += c * v.x; a1 += c * v.y; a2 += c * v.z; a3 += c * v.w;
    }
    float4 o; o.x = a0; o.y = a1; o.z = a2; o.w = a3;
    ((float4*)(hout + (size_t)i * HIDDEN))[lane] = o;
}

// ------------------------- GEMM2: logits = h*W2^T + b2 -> d_out (cols 0..39)
__global__ __launch_bounds__(64) void k_gemm2(const float* __restrict__ h,
                                              const float* __restrict__ W2,
                                              const float* __restrict__ b2,
                                              float* __restrict__ out) {
    const int nt = blockIdx.x * 2 + (threadIdx.x >> 5);
    if (nt >= NODE_TILES) return;                  // wave-uniform: EXEC stays all-1s
    const int lane = threadIdx.x & 31;
    const int l15  = lane & 15;
    const int hi   = lane >> 4;

    const float* hrow = h + (size_t)(nt * 16 + l15) * HIDDEN;
    v8f acc0 = {}, acc1 = {}, acc2 = {};
    for (int k = 0; k < HIDDEN; k += 4) {
        const int kb = k + 2 * hi;
        v2f a = *(const v2f*)(hrow + kb);
        const int c2 = 32 + l15;
        v2f b0  = *(const v2f*)(W2 + (size_t)l15 * HIDDEN + kb);
        v2f b1v = *(const v2f*)(W2 + (size_t)(16 + l15) * HIDDEN + kb);
        v2f b2v; b2v.x = 0.0f; b2v.y = 0.0f;
        if (c2 < NUM_CLASS) b2v = *(const v2f*)(W2 + (size_t)c2 * HIDDEN + kb);
        acc0 = __builtin_amdgcn_wmma_f32_16x16x4_f32(false, a, false, b0,  (short)0, acc0, false, false);
        acc1 = __builtin_amdgcn_wmma_f32_16x16x4_f32(false, a, false, b1v, (short)0, acc1, false, false);
        acc2 = __builtin_amdgcn_wmma_f32_16x16x4_f32(false, a, false, b2v, (short)0, acc2, false, false);
    }
#pragma unroll
    for (int r = 0; r < 8; ++r) {
        const size_t orow = (size_t)(nt * 16 + r + hi * 8) * NUM_CLASS;
        out[orow + l15]      = acc0[r] + b2[l15];
        out[orow + l15 + 16] = acc1[r] + b2[l15 + 16];
        if (l15 + 32 < NUM_CLASS) out[orow + l15 + 32] = acc2[r] + b2[l15 + 32];
    }
}

// ------------------------------------------ in-place log_softmax (1 wave/node)
__global__ __launch_bounds__(256) void k_lsm(float* __restrict__ out) {
    const int n    = (blockIdx.x * 256 + threadIdx.x) >> 5;
    const int lane = threadIdx.x & 31;
    if (n >= N_NODES) return;
    float* o = out + (size_t)n * NUM_CLASS;
    const float v0 = o[lane];                                             // classes 0..31
    const float v1 = (lane < NUM_CLASS - 32) ? o[32 + lane] : -INFINITY;  // 32..39
    float m = fmaxf(v0, v1);
#pragma unroll
    for (int s = 16; s >= 1; s >>= 1) m = fmaxf(m, __shfl_xor(m, s, 32));
    float sum = __expf(v0 - m) + ((lane < NUM_CLASS - 32) ? __expf(v1 - m) : 0.0f);
#pragma unroll
    for (int s = 16; s >= 1; s >>= 1) sum += __shfl_xor(sum, s, 32);
    const float lg = m + __logf(sum);
    o[lane] = v0 - lg;
    if (lane < NUM_CLASS - 32) o[32 + lane] = v1 - lg;
}

// ---------------------------------------------------------------- launcher
extern "C" void kernel_launch(void* const* d_in, const int* in_sizes, int n_in,
                              void* d_out, int out_size, void* d_ws, size_t ws_size,
                              hipStream_t stream) {
    const float* x    = (const float*)d_in[0];
    const int*   ei   = (const int*)d_in[1];      // [2, E] flat: src row then dst row
    const float* W1   = (const float*)d_in[2];    // [128, 512]
    const float* b1   = (const float*)d_in[3];
    const float* W2   = (const float*)d_in[4];    // [40, 128]
    const float* b2   = (const float*)d_in[5];
    const float* attl = (const float*)d_in[6];    // [4, 128]
    const float* attr = (const float*)d_in[7];
    float* out = (float*)d_out;

    char*  ws  = (char*)d_ws;
    size_t off = 0;
    auto alloc = [&](size_t bytes) -> void* {
        void* p = ws + off;
        off += (bytes + 255) & ~(size_t)255;
        return p;
    };
    const size_t hbytes = sizeof(float) * (size_t)N_NODES * HIDDEN;
    float* raw      = (float*)alloc(hbytes);
    float* bufA     = (float*)alloc(hbytes);
    float* bufB     = (float*)alloc(hbytes);
    int*   cnt      = (int*)  alloc(sizeof(int)   * N_NODES);
    int*   rowptr   = (int*)  alloc(sizeof(int)   * (N_NODES + 1));
    int*   partial  = (int*)  alloc(sizeof(int)   * N_NODES);
    int*   blocksum = (int*)  alloc(sizeof(int)   * SCAN_NB);
    int*   cursor   = (int*)  alloc(sizeof(int)   * N_NODES);
    int*   csr_src  = (int*)  alloc(sizeof(int)   * N_EDGES);
    int*   csr_dst  = (int*)  alloc(sizeof(int)   * N_EDGES);
    float* csr_norm = (float*)alloc(sizeof(float) * N_EDGES);
    float* coef     = (float*)alloc(sizeof(float) * N_EDGES);
    float* dis      = (float*)alloc(sizeof(float) * N_NODES);
    float* al       = (float*)alloc(sizeof(float) * N_NODES);
    float* ar       = (float*)alloc(sizeof(float) * N_NODES);
    (void)in_sizes; (void)n_in; (void)out_size; (void)ws_size;

    const int nodeBlocks = (N_NODES + 255) / 256;
    const int edgeBlocks = (N_EDGES + 255) / 256;

    // ---- one-time CSR build (by destination) + normalization
    hipMemsetAsync(cnt, 0, sizeof(int) * N_NODES, stream);
    k_count<<<edgeBlocks, 256, 0, stream>>>(ei, cnt);
    k_scan1<<<SCAN_NB, SCAN_B, 0, stream>>>(cnt, partial, blocksum);
    k_scan2<<<1, SCAN_B, 0, stream>>>(blocksum);
    k_scan3<<<SCAN_NB, SCAN_B, 0, stream>>>(partial, blocksum, rowptr);
    k_dis_cursor<<<nodeBlocks, 256, 0, stream>>>(cnt, rowptr, dis, cursor);
    k_fill<<<edgeBlocks, 256, 0, stream>>>(ei, dis, cursor, csr_src, csr_dst, csr_norm);

    // ---- input MLP (WMMA fp32, LDS-staged A tile)
    k_gemm1<<<NODE_TILES, 128, 0, stream>>>(x, W1, b1, raw);

    // ---- FAConv layers: coef pass + pull-mode gather, no fp32 atomics
    const float* hcur = raw;
    float* pp[2] = { bufA, bufB };
    for (int l = 0; l < NUM_LAYERS; ++l) {
        float* hnext = pp[l & 1];
        k_att <<<(N_NODES * 32 + 255) / 256, 256, 0, stream>>>(hcur, attl + l * HIDDEN, attr + l * HIDDEN, al, ar);
        k_coef<<<edgeBlocks, 256, 0, stream>>>(csr_src, csr_dst, csr_norm, al, ar, coef);
        k_agg <<<(N_NODES * 32 + 255) / 256, 256, 0, stream>>>(rowptr, csr_src, coef,
                                                               hcur, raw, al, ar, dis, hnext);
        hcur = hnext;
    }

    // ---- classifier + log_softmax (WMMA fp32, masked to 40 classes)
    k_gemm2<<<(NODE_TILES + 1) / 2, 64, 0, stream>>>(hcur, W2, b2, out);
    k_lsm  <<<(N_NODES * 32 + 255) / 256, 256, 0, stream>>>(out);
}